// cluster_embedding_loss_446676599062
// MI455X (gfx1250) — compile-verified
//
#include <hip/hip_runtime.h>
#include <hip/hip_bf16.h>
#include <math.h>

// Problem constants (from reference)
static constexpr int BS   = 8;
static constexpr int NV   = 50000;
static constexpr int D    = 128;
static constexpr int KCL  = 50;
static constexpr float EPSN = 1e-12f;

typedef __attribute__((ext_vector_type(2))) float v2f;
typedef __attribute__((ext_vector_type(8))) float v8f;

// One block per (b,k) segment. 256 threads = 8 waves; wave w owns D-columns
// [16w, 16w+16). WMMA f32 16x16x4 with B = ones accumulates per-column
// row-sums of x and x^2 across the segment.
__global__ __launch_bounds__(256) void seg_stats_kernel(
    const float* __restrict__ emb,     // (BS, NV, D)
    const int*   __restrict__ sizes,   // (BS, K)
    float* __restrict__ mean_n,        // (BS*K, D) unit mean vectors
    float* __restrict__ msq,           // (BS*K)   ||mean_n||^2
    float* __restrict__ varsum)        // (BS*K)   sum_d var_d
{
    const int bk = blockIdx.x;          // 0..BS*K-1
    const int b  = bk / KCL;
    const int k  = bk % KCL;
    const int n  = sizes[b * KCL + k];
    const int start = (k == 0) ? 0 : sizes[b * KCL + k - 1];
    const float* base = emb + ((size_t)b * NV + (size_t)start) * D;

    const int tid   = threadIdx.x;
    const int wave  = tid >> 5;         // 0..7
    const int lane  = tid & 31;
    const int dbase = wave * 16;        // this wave's 16 D-columns
    const int half  = lane >> 4;        // 0: rows j+0/j+1, 1: rows j+2/j+3
    const int l     = lane & 15;        // M index within the half
    const int r0    = half * 2;         // K row offset for a.x
    const int col   = dbase + l;

    v8f c1 = {0.f,0.f,0.f,0.f,0.f,0.f,0.f,0.f};   // sum x
    v8f c2 = {0.f,0.f,0.f,0.f,0.f,0.f,0.f,0.f};   // sum x^2
    v2f bones; bones.x = 1.0f; bones.y = 1.0f;    // B layout irrelevant (all ones)

    for (int j = 0; j < n; j += 4) {
        const int ja = j + r0;
        const int jb = ja + 1;
        // Unconditional loads (always within the embeddings array); mask to 0
        // for rows past the segment end so they contribute nothing.
        float xa = base[(size_t)ja * D + col];
        float xb = base[(size_t)jb * D + col];
        xa = (ja < n) ? xa : 0.0f;
        xb = (jb < n) ? xb : 0.0f;
        v2f a;  a.x  = xa;      a.y  = xb;
        v2f a2; a2.x = xa * xa; a2.y = xb * xb;
        // D = A x ones + C : per-column row sums, accumulated across the loop.
        c1 = __builtin_amdgcn_wmma_f32_16x16x4_f32(
                 false, a,  false, bones, (short)0, c1, false, false);
        c2 = __builtin_amdgcn_wmma_f32_16x16x4_f32(
                 false, a2, false, bones, (short)0, c2, false, false);
    }

    // C/D 16x16 f32 layout: VGPR r, lanes 0-15 hold M=r; lanes 16-31 hold M=r+8.
    // All N columns identical (B was ones) -> lane 0 / lane 16 spill to LDS.
    __shared__ float s1[D];
    __shared__ float s2[D];
    if (lane == 0) {
#pragma unroll
        for (int r = 0; r < 8; ++r) { s1[dbase + r] = c1[r]; s2[dbase + r] = c2[r]; }
    }
    if (lane == 16) {
#pragma unroll
        for (int r = 0; r < 8; ++r) { s1[dbase + 8 + r] = c1[r]; s2[dbase + 8 + r] = c2[r]; }
    }
    __syncthreads();

    // Threads 0..127: one D-column each. var_d = (s2 - s1^2/n)/(n-1).
    __shared__ float redm[D];
    __shared__ float redv[D];
    __shared__ float sinv;
    const float fn = (float)n;
    float meanv = 0.0f;
    if (tid < D) {
        const float a1 = s1[tid];
        const float a2 = s2[tid];
        meanv = a1 / fn;
        const float var = (a2 - a1 * a1 / fn) / (fn - 1.0f);
        redm[tid] = meanv * meanv;
        redv[tid] = var;
    }
    __syncthreads();
#pragma unroll
    for (int st = 64; st > 0; st >>= 1) {
        if (tid < st) { redm[tid] += redm[tid + st]; redv[tid] += redv[tid + st]; }
        __syncthreads();
    }
    if (tid == 0) {
        const float norm2 = redm[0];
        const float nrm   = sqrtf(norm2);
        const float m     = fmaxf(nrm, EPSN);
        msq[bk]    = norm2 / (m * m);   // ||mean_n||^2 (==1 unless mean ~ 0)
        varsum[bk] = redv[0];
        sinv = 1.0f / m;
    }
    __syncthreads();
    if (tid < D) {
        mean_n[(size_t)bk * D + tid] = meanv * sinv;
    }
}

// Single block, 128 threads (one per D-column). Tiny: 8*50*128 reads.
__global__ __launch_bounds__(128) void finalize_kernel(
    const float* __restrict__ mean_n,
    const float* __restrict__ msq,
    const float* __restrict__ varsum,
    float* __restrict__ out)
{
    const int tid = threadIdx.x;     // D-column
    __shared__ float red[D];
    __shared__ float total_s;
    if (tid == 0) total_s = 0.0f;
    const float npairs = (float)KCL * (KCL - 1) * 0.5f;

    for (int b = 0; b < BS; ++b) {
        float s = 0.0f;
        for (int k = 0; k < KCL; ++k)
            s += mean_n[((size_t)(b * KCL + k)) * D + tid];
        red[tid] = s * s;
        __syncthreads();
#pragma unroll
        for (int st = 64; st > 0; st >>= 1) {
            if (tid < st) red[tid] += red[tid + st];
            __syncthreads();
        }
        if (tid == 0) {
            const float sq_of_sum = red[0];
            float sum_of_sq = 0.0f, lv = 0.0f;
            for (int k = 0; k < KCL; ++k) {
                sum_of_sq += msq[b * KCL + k];
                lv        += varsum[b * KCL + k];
            }
            const float loss_dist = npairs + 0.5f * (sq_of_sum - sum_of_sq);
            total_s += 0.1f * loss_dist + lv;
        }
        __syncthreads();
    }
    if (tid == 0) out[0] = total_s;
}

extern "C" void kernel_launch(void* const* d_in, const int* in_sizes, int n_in,
                              void* d_out, int out_size, void* d_ws, size_t ws_size,
                              hipStream_t stream) {
    const float* emb   = (const float*)d_in[0];   // (BS, NV, D) fp32
    const int*   sizes = (const int*)d_in[1];     // (BS, K) int32

    float* ws      = (float*)d_ws;
    float* mean_n  = ws;                              // BS*K*D floats
    float* msq     = mean_n + (size_t)BS * KCL * D;   // BS*K floats
    float* varsum  = msq + (size_t)BS * KCL;          // BS*K floats

    seg_stats_kernel<<<BS * KCL, 256, 0, stream>>>(emb, sizes, mean_n, msq, varsum);
    finalize_kernel<<<1, 128, 0, stream>>>(mean_n, msq, varsum, (float*)d_out);
}